// VessPro_7395933684028
// MI455X (gfx1250) — compile-verified
//
#include <hip/hip_runtime.h>

typedef __attribute__((ext_vector_type(2))) float v2f;
typedef __attribute__((ext_vector_type(8))) float v8f;
typedef __attribute__((ext_vector_type(4))) unsigned int u32x4;
typedef __attribute__((ext_vector_type(8))) int i32x8;
typedef __attribute__((ext_vector_type(4))) int i32x4;

#define NB 4
#define CH 320
#define LL 4096
#define TM 32
#define TN 64
#define KC 16

__global__ void vess_init_ws(float* ws) {
    if (threadIdx.x < 16) ws[threadIdx.x] = 0.0f;
}

// Issue one TDM 2D tile load: tile_dim1 rows x tile_dim0 f32 elements,
// row stride `stride0` elements in global memory, +1 dword LDS pad per row.
// pad_interval_enc: 4 -> pad every 32 dwords, 5 -> every 64 dwords.
__device__ __forceinline__ void tdm_load_2d(unsigned lds_byte_off,
                                            const void* gaddr,
                                            unsigned tile_dim0, unsigned tile_dim1,
                                            unsigned tensor_dim0, unsigned tensor_dim1,
                                            unsigned stride0,
                                            unsigned pad_interval_enc)
{
    unsigned long long ga = (unsigned long long)(uintptr_t)gaddr;
    u32x4 g0;
    g0[0] = 1u;                                   // count=1, user descriptor, no gather
    g0[1] = lds_byte_off;                         // LDS dest (bytes)
    g0[2] = (unsigned)(ga & 0xffffffffu);         // global_addr lo
    g0[3] = (unsigned)((ga >> 32) & 0x1ffffffu)   // global_addr hi (57-bit)
          | 0x80000000u;                          // type = 2 ("image")
    i32x8 g1;
    g1[0] = (int)((2u << 16)                      // data_size = 4 bytes
          | (1u << 20)                            // pad_enable
          | (pad_interval_enc << 22));            // pad_amount enc 0 => 1 dword
    g1[1] = (int)((tensor_dim0 & 0xffffu) << 16);               // tensor_dim0 lo16
    g1[2] = (int)(((tensor_dim0 >> 16) & 0xffffu)
          | ((tensor_dim1 & 0xffffu) << 16));                   // dim0 hi | dim1 lo
    g1[3] = (int)(((tensor_dim1 >> 16) & 0xffffu)
          | ((tile_dim0 & 0xffffu) << 16));                     // dim1 hi | tile_dim0
    g1[4] = (int)(tile_dim1 & 0xffffu);                         // tile_dim1, tile_dim2=0
    g1[5] = (int)stride0;                                       // tensor_dim0_stride lo32
    g1[6] = 0;                                                  // stride0 hi, stride1 lo
    g1[7] = 0;
    i32x4 z4 = {0, 0, 0, 0};
    i32x8 z8 = {0, 0, 0, 0, 0, 0, 0, 0};
    __builtin_amdgcn_tensor_load_to_lds(g0, g1, z4, z4, z8, 0);
}

// One block: 8 waves, 32x64 output tile, wave (wm,wn) owns a 16x16 WMMA tile.
// z = branch*4 + n.  branch 0: q batch = n;  branch 1 (flipped): q batch = 3-n.
__global__ void __launch_bounds__(256) vess_gemm_mask(
    const float* __restrict__ q, const float* __restrict__ k,
    const float* __restrict__ coord_q, const float* __restrict__ coord_k,
    float* __restrict__ ws)
{
    // double-buffered, rows padded +1 dword (TDM pad reproduces this)
    __shared__ float qs[2][KC][TM + 1];
    __shared__ float ks[2][KC][TN + 1];

    const int z = blockIdx.z;
    const int branch = z >> 2;
    const int n  = z & 3;
    const int nq = branch ? (NB - 1 - n) : n;

    const float* __restrict__ qb = q + (size_t)nq * CH * LL;
    const float* __restrict__ kb = k + (size_t)n  * CH * LL;

    const int l0 = blockIdx.x * TM;
    const int m0 = blockIdx.y * TN;

    const int tid  = threadIdx.x;
    const int lane = tid & 31;
    const int wave = tid >> 5;
    const int half = lane >> 4;     // 0: lanes 0-15, 1: lanes 16-31
    const int ln   = lane & 15;
    const int wm   = wave >> 2;     // 0..1
    const int wn   = wave & 3;      // 0..3

    const unsigned qs0_off = (unsigned)(uintptr_t)(void*)&qs[0][0][0];
    const unsigned qs1_off = (unsigned)(uintptr_t)(void*)&qs[1][0][0];
    const unsigned ks0_off = (unsigned)(uintptr_t)(void*)&ks[0][0][0];
    const unsigned ks1_off = (unsigned)(uintptr_t)(void*)&ks[1][0][0];

    v8f acc = {};

    const int nchunks = CH / KC;    // 20

    // prologue: DMA chunk 0 into buffer 0 (wave 0 only; TDM ignores EXEC)
    if (wave == 0) {
        tdm_load_2d(qs0_off, qb + l0, TM, KC, LL, CH, LL, 4);
        tdm_load_2d(ks0_off, kb + m0, TN, KC, LL, CH, LL, 5);
    }

    for (int ci = 0; ci < nchunks; ++ci) {
        const int buf = ci & 1;
        if (wave == 0) {
            if (ci + 1 < nchunks) {
                const size_t row = (size_t)(ci + 1) * KC * LL;
                tdm_load_2d(buf ? qs0_off : qs1_off, qb + row + l0, TM, KC, LL, CH, LL, 4);
                tdm_load_2d(buf ? ks0_off : ks1_off, kb + row + m0, TN, KC, LL, CH, LL, 5);
                __builtin_amdgcn_s_wait_tensorcnt(2);  // chunk ci complete; ci+1 in flight
            } else {
                __builtin_amdgcn_s_wait_tensorcnt(0);
            }
        }
        __syncthreads();   // LDS buffer `buf` ready for all waves

        // ---- 4 x V_WMMA_F32_16X16X4_F32 over this K-chunk ----
        // A layout (16x4 f32): lane(half,m), VGPR v holds K = 2*half + v
        // B layout (4x16 f32): lane(half,n), VGPR v holds K = 2*half + v
        const float (* __restrict__ qsb)[TM + 1] = qs[buf];
        const float (* __restrict__ ksb)[TN + 1] = ks[buf];
        #pragma unroll
        for (int k4 = 0; k4 < KC; k4 += 4) {
            v2f a, b;
            a.x = qsb[k4 + 2 * half    ][wm * 16 + ln];
            a.y = qsb[k4 + 2 * half + 1][wm * 16 + ln];
            b.x = ksb[k4 + 2 * half    ][wn * 16 + ln];
            b.y = ksb[k4 + 2 * half + 1][wn * 16 + ln];
            acc = __builtin_amdgcn_wmma_f32_16x16x4_f32(
                false, a, false, b, (short)0, acc, false, false);
        }
        __syncthreads();   // done reading `buf`; wave 0 may DMA over it next iter
    }

    // ---- fused mask + reduction epilogue (logit matrix never hits memory) ----
    const float* cqn = coord_q + nq * 6;  // q-centers box (flipped batch for branch 1)
    const float* ckn = coord_k + n  * 6;
    const float* cqd = coord_q + n  * 6;  // max_diag uses UNflipped coord_q (closure capture)

    const float inv16 = 1.0f / 16.0f;
    float bq0 = (cqn[3] - cqn[0]) * inv16;
    float bq1 = (cqn[4] - cqn[1]) * inv16;
    float bq2 = (cqn[5] - cqn[2]) * inv16;
    float bk0 = (ckn[3] - ckn[0]) * inv16;
    float bk1 = (ckn[4] - ckn[1]) * inv16;
    float bk2 = (ckn[5] - ckn[2]) * inv16;
    float dq0 = (cqd[3] - cqd[0]) * inv16;
    float dq1 = (cqd[4] - cqd[1]) * inv16;
    float dq2 = (cqd[5] - cqd[2]) * inv16;

    float diag_q = sqrtf(dq0 * dq0 + dq1 * dq1 + dq2 * dq2);
    float diag_k = sqrtf(bk0 * bk0 + bk1 * bk1 + bk2 * bk2);
    float md = fmaxf(diag_q, diag_k);
    float thr2 = 0.25f * md * md;   // dist/md < 0.5  <=>  d2 < (0.5*md)^2

    // k-center for this thread's output column (constant over r)
    int m = m0 + wn * 16 + ln;
    float ckx = (float)((m >> 8)     ) * bk0 + 0.5f * bk0 + ckn[0];
    float cky = (float)((m >> 4) & 15) * bk1 + 0.5f * bk1 + ckn[1];
    float ckz = (float)( m       & 15) * bk2 + 0.5f * bk2 + ckn[2];

    float num = 0.0f, cnt = 0.0f;
    // C/D layout: VGPR r holds element (M = r + 8*half, N = ln)
    #pragma unroll
    for (int r = 0; r < 8; ++r) {
        int l = l0 + wm * 16 + r + 8 * half;
        float qx = (float)((l >> 8)     ) * bq0 + 0.5f * bq0 + cqn[0];
        float qy = (float)((l >> 4) & 15) * bq1 + 0.5f * bq1 + cqn[1];
        float qz = (float)( l       & 15) * bq2 + 0.5f * bq2 + cqn[2];
        float dx = qx - ckx, dy = qy - cky, dz = qz - ckz;
        float d2 = dx * dx + dy * dy + dz * dz;
        if (d2 < thr2) { num += acc[r]; cnt += 1.0f; }
    }

    // wave32 shuffle reduction, one atomic pair per wave
    #pragma unroll
    for (int off = 16; off; off >>= 1) {
        num += __shfl_xor(num, off, 32);
        cnt += __shfl_xor(cnt, off, 32);
    }
    if (lane == 0) {
        atomicAdd(&ws[z * 2 + 0], num);
        atomicAdd(&ws[z * 2 + 1], cnt);
    }
}

__global__ void vess_finalize(const float* __restrict__ ws, float* __restrict__ out) {
    if (threadIdx.x == 0 && blockIdx.x == 0) {
        float s = 0.0f;
        #pragma unroll
        for (int zz = 0; zz < 8; ++zz)
            s += ws[2 * zz] / (ws[2 * zz + 1] + 1e-6f);
        // -2*mean over N=4 for each branch == -0.5 * sum over all 8 (branch,batch) pairs
        out[0] = -0.5f * s;
    }
}

extern "C" void kernel_launch(void* const* d_in, const int* in_sizes, int n_in,
                              void* d_out, int out_size, void* d_ws, size_t ws_size,
                              hipStream_t stream) {
    const float* q  = (const float*)d_in[0];
    const float* k  = (const float*)d_in[1];
    const float* cq = (const float*)d_in[2];
    const float* ck = (const float*)d_in[3];
    float* ws  = (float*)d_ws;
    float* out = (float*)d_out;

    vess_init_ws<<<1, 32, 0, stream>>>(ws);
    dim3 grid(LL / TM, LL / TN, 8);   // 128 x 64 x 8
    vess_gemm_mask<<<grid, 256, 0, stream>>>(q, k, cq, ck, ws);
    vess_finalize<<<1, 1, 0, stream>>>(ws, out);
}